// TrainableODE_82824149336590
// MI455X (gfx1250) — compile-verified
//
#include <hip/hip_runtime.h>

#define N_TS   1001
#define NSTEPS (N_TS - 1)
#define BATCH  4096
#define NC     6

typedef float v2f __attribute__((ext_vector_type(2)));

struct SmithParams {
    float Ees_lv, Ees_rv, P0_lv, P0_rv, lam_lv, lam_rv;
    float E_ao, E_vc, E_pa, E_pu;
    float iR_mt, iR_av, iR_sys, iR_tc, iR_pv, iR_pul;
    float A, negB, C, T;
};

// e(t) = A * exp(-B * (mod(t,T) - C)^2)
// t >= 0, T > 0  ->  fmodf == jnp.mod, and fmod is an *exact* remainder,
// so this bit-matches the reference's wrap decision.
__device__ __forceinline__ float activation(float t, const SmithParams& p) {
    float m = fmodf(t, p.T) - p.C;
    return p.A * __expf(p.negB * m * m);
}

// dV = f(e, V). exp(x)-1 replaces expm1: args ~2..3.3, no cancellation.
__device__ __forceinline__ void vfield(float e, const float V[NC],
                                       const SmithParams& p, float dV[NC]) {
    const float ome  = 1.0f - e;
    const float P_lv = e * p.Ees_lv * V[0] + ome * p.P0_lv * (__expf(p.lam_lv * V[0]) - 1.0f);
    const float P_rv = e * p.Ees_rv * V[3] + ome * p.P0_rv * (__expf(p.lam_rv * V[3]) - 1.0f);
    const float P_ao = p.E_ao * V[1];
    const float P_vc = p.E_vc * V[2];
    const float P_pa = p.E_pa * V[4];
    const float P_pu = p.E_pu * V[5];

    const float Q_mt  = fmaxf(P_pu - P_lv, 0.0f) * p.iR_mt;
    const float Q_av  = fmaxf(P_lv - P_ao, 0.0f) * p.iR_av;
    const float Q_sys = (P_ao - P_vc) * p.iR_sys;
    const float Q_tc  = fmaxf(P_vc - P_rv, 0.0f) * p.iR_tc;
    const float Q_pv  = fmaxf(P_rv - P_pa, 0.0f) * p.iR_pv;
    const float Q_pul = (P_pa - P_pu) * p.iR_pul;

    dV[0] = Q_mt  - Q_av;
    dV[1] = Q_av  - Q_sys;
    dV[2] = Q_sys - Q_tc;
    dV[3] = Q_tc  - Q_pv;
    dV[4] = Q_pv  - Q_pul;
    dV[5] = Q_pul - Q_mt;
}

__device__ __forceinline__ void store_row_nt(float* __restrict__ dst, const float V[NC]) {
    // dst is 8-byte aligned for every member (24B stride). Three NT b64 stores,
    // coalesced across the 32 lanes of the wave.
    v2f a; a.x = V[0]; a.y = V[1];
    v2f b; b.x = V[2]; b.y = V[3];
    v2f c; c.x = V[4]; c.y = V[5];
    __builtin_nontemporal_store(a, (v2f*)(dst + 0));
    __builtin_nontemporal_store(b, (v2f*)(dst + 2));
    __builtin_nontemporal_store(c, (v2f*)(dst + 4));
}

__global__ __launch_bounds__(32)
void smith_heun_kernel(const float* __restrict__ ts,
                       const float* __restrict__ y0,
                       const float* __restrict__ params,
                       float* __restrict__ out) {
    // grid * block == BATCH exactly: no bounds check needed.
    const int b = blockIdx.x * 32 + threadIdx.x;

    SmithParams p;
    p.Ees_lv = params[0];  p.Ees_rv = params[1];
    p.P0_lv  = params[2];  p.P0_rv  = params[3];
    p.lam_lv = params[4];  p.lam_rv = params[5];
    p.E_ao   = params[6];  p.E_vc   = params[7];
    p.E_pa   = params[8];  p.E_pu   = params[9];
    p.iR_mt  = 1.0f / params[10];
    p.iR_av  = 1.0f / params[11];
    p.iR_sys = 1.0f / params[12];
    p.iR_tc  = 1.0f / params[13];
    p.iR_pv  = 1.0f / params[14];
    p.iR_pul = 1.0f / params[15];
    p.A      = params[16];
    p.negB   = -params[17];
    p.C      = params[18];
    p.T      = params[19];

    const float dt  = ts[1] - ts[0];   // == 0.01f; (float)k * dt is bit-identical to ts[k]
    const float hdt = 0.5f * dt;

    float V[NC];
    #pragma unroll
    for (int i = 0; i < NC; ++i) V[i] = y0[b * NC + i];

    // Running output pointer: one 64-bit add per row instead of a mul chain.
    float* orow = out + (size_t)b * NC;
    store_row_nt(orow, V);            // row 0 = y0

    #pragma clang loop unroll_count(2)
    for (int k = 0; k < NSTEPS; ++k) {
        const float t  = (float)k * dt;
        // state-independent: scheduler can hoist these off the V-chain
        const float e1 = activation(t, p);
        const float e2 = activation(t + dt, p);   // reference uses t+dt, not ts[k+1]

        float k1[NC], k2[NC], ym[NC];
        vfield(e1, V, p, k1);
        #pragma unroll
        for (int i = 0; i < NC; ++i) ym[i] = V[i] + dt * k1[i];
        vfield(e2, ym, p, k2);
        #pragma unroll
        for (int i = 0; i < NC; ++i) V[i] = V[i] + hdt * (k1[i] + k2[i]);

        orow += (size_t)BATCH * NC;
        store_row_nt(orow, V);
    }
}

extern "C" void kernel_launch(void* const* d_in, const int* in_sizes, int n_in,
                              void* d_out, int out_size, void* d_ws, size_t ws_size,
                              hipStream_t stream) {
    (void)in_sizes; (void)n_in; (void)d_ws; (void)ws_size; (void)out_size;
    const float* ts     = (const float*)d_in[0];   // [1001]
    const float* y0     = (const float*)d_in[1];   // [4096, 6]
    const float* params = (const float*)d_in[2];   // [20]
    float*       out    = (float*)d_out;           // [1001, 4096, 6]

    dim3 block(32);                 // one wave32 per workgroup -> max WGP spread
    dim3 grid(BATCH / 32);          // 128 single-wave workgroups
    smith_heun_kernel<<<grid, block, 0, stream>>>(ts, y0, params, out);
}